// NeuralNet_13829794693862
// MI455X (gfx1250) — compile-verified
//
#include <hip/hip_runtime.h>

// Problem constants (match reference)
#define NMENT 8192
#define NC    30
#define DDIM  300      // true K
#define KPAD  320      // K padded to 8 waves * 10 steps * 4
#define LSTR  324      // LDS row stride in floats: 324 mod 64 == 4 -> conflict-free b64
#define WINSZ 100
#define TOPK  25
#define NEGV  (-1e10f)

typedef __attribute__((ext_vector_type(2))) float v2f;
typedef __attribute__((ext_vector_type(8))) float v8f;

// One workgroup (256 threads = 8 waves) per mention.
__global__ __launch_bounds__(256, 1)
void mention_fused_kernel(const float* __restrict__ emb,        // (N, NC, D)
                          const float* __restrict__ tok,        // (N, WIN, D)
                          const unsigned char* __restrict__ msk,// (N, WIN) bool (1B)
                          const float* __restrict__ Bd1,        // (D,)
                          const float* __restrict__ Bd2,        // (D,)
                          float* __restrict__ out)              // (N, NC)
{
    const int n    = blockIdx.x;
    const int tid  = threadIdx.x;
    const int lane = tid & 31;
    const int wid  = tid >> 5;
    const int lo   = lane & 15;   // fragment row/col index
    const int hi   = lane >> 4;   // fragment K-half selector

    __shared__ float emb_s[32 * LSTR];   // A: 32 cands x [0,KPAD) (rows>=30, cols>=300 zero)
    __shared__ float tok_s[16 * LSTR];   // B: 16 tokens x [0,KPAD), scaled by B_diag2
    __shared__ float scores_s[32 * 16];  // C tile: 32 cands x 16 tokens
    __shared__ float tscore_s[128];      // token max-scores, padded to 128

    // small late-phase buffers alias tok_s (dead after the chunk loop)
    float* fcs_s   = tok_s;              // [0..299]
    float* probs_s = tok_s + 320;        // [320..351]
    int*   besti_s = (int*)(tok_s + 352);// [352..383]

    const float*  embN  = emb + (size_t)n * NC * DDIM;
    const float*  tokN  = tok + (size_t)n * WINSZ * DDIM;
    const float4* embN4 = (const float4*)embN;
    const float4* tokN4 = (const float4*)tokN;
    const float4* bd2v  = (const float4*)Bd2;
    const float4  zero4 = make_float4(0.f, 0.f, 0.f, 0.f);

    // ---- stage embeddings into LDS as float4 (zero-pad cands 30,31 and K [300,320)) ----
    for (int i = tid; i < 32 * 75; i += 256) {           // 2400 float4 tiles
        int c = i / 75, d4 = i - c * 75;
        float4 v = (i < NC * 75) ? embN4[i] : zero4;     // linear index == c*75+d4
        ((float4*)(emb_s + c * LSTR))[d4] = v;
    }
    for (int i = tid; i < 32 * 5; i += 256) {            // K pad columns [300,320)
        int r = i / 5, c4 = i - r * 5;
        ((float4*)(emb_s + r * LSTR))[75 + c4] = zero4;
    }
    if (tid < 128) tscore_s[tid] = NEGV;                 // pads [100..127] stay NEG
    __syncthreads();

    // ---- phase 1: scores = emb x (tok * Bd2)^T via WMMA, 7 chunks of 16 tokens ----
    for (int chunk = 0; chunk < 7; ++chunk) {
        // stage one 16-token chunk (scaled by B_diag2), float4 wide
        for (int j = tid; j < 16 * 75; j += 256) {
            int wl = j / 75, d4 = j - wl * 75;
            int w  = chunk * 16 + wl;
            float4 v = zero4;
            if (w < WINSZ) {
                float4 t = tokN4[w * 75 + d4];
                float4 s = bd2v[d4];
                v = make_float4(t.x * s.x, t.y * s.y, t.z * s.z, t.w * s.w);
            }
            ((float4*)(tok_s + wl * LSTR))[d4] = v;
        }
        for (int j = tid; j < 16 * 5; j += 256) {        // K pad columns
            int wl = j / 5, c4 = j - wl * 5;
            ((float4*)(tok_s + wl * LSTR))[75 + c4] = zero4;
        }
        for (int j = tid; j < 32 * 16; j += 256) scores_s[j] = 0.0f;
        __syncthreads();

        // prefetch next chunk's tokens (global_prefetch_b8) to hide HBM latency
        if (chunk + 1 < 7) {
            const float* nxt = tokN + (size_t)(chunk + 1) * 16 * DDIM;
            for (int p = tid; p < 150; p += 256)          // 150 x 128B lines = 19.2 KB
                __builtin_prefetch(nxt + p * 32, 0, 1);
        }

        // each wave owns a contiguous block of 10 K4-steps (literal trip count)
        v8f c0 = {}; v8f c1 = {};
        const int kwave = wid * 40;                      // 10 steps * K=4
        #pragma unroll
        for (int i = 0; i < 10; ++i) {
            const int kb = kwave + 4 * i + 2 * hi;
            v2f a0, a1, b;
            a0.x = emb_s[lo * LSTR + kb];        a0.y = emb_s[lo * LSTR + kb + 1];
            a1.x = emb_s[(lo + 16) * LSTR + kb]; a1.y = emb_s[(lo + 16) * LSTR + kb + 1];
            b.x  = tok_s[lo * LSTR + kb];        b.y  = tok_s[lo * LSTR + kb + 1];
            c0 = __builtin_amdgcn_wmma_f32_16x16x4_f32(false, a0, false, b,
                                                       (short)0, c0, false, false);
            c1 = __builtin_amdgcn_wmma_f32_16x16x4_f32(false, a1, false, b,
                                                       (short)0, c1, false, false);
        }
        // cross-wave reduce partial C tiles into LDS (ds_add_f32)
        #pragma unroll
        for (int v = 0; v < 8; ++v) {
            atomicAdd(&scores_s[(v + 8 * hi) * 16 + lo],      c0[v]);
            atomicAdd(&scores_s[(16 + v + 8 * hi) * 16 + lo], c1[v]);
        }
        __syncthreads();

        // masked column-max over candidates -> token_scores
        if (tid < 16) {
            int w = chunk * 16 + tid;
            if (w < WINSZ) {
                float m = scores_s[tid];
                for (int c = 1; c < NC; ++c)
                    m = fmaxf(m, scores_s[c * 16 + tid]);
                tscore_s[w] = msk[(size_t)n * WINSZ + w] ? m : NEGV;
            }
        }
        __syncthreads();
    }

    // ---- phase 2: top-25 (iterative argmax, register-resident) + softmax (wave 0) ----
    if (wid == 0) {
        float v[4];
        #pragma unroll
        for (int j = 0; j < 4; ++j) v[j] = tscore_s[lane + 32 * j];

        float myVal = -3.402823466e38f;
        int   myIdx = 0;
        for (int it = 0; it < TOPK; ++it) {
            float bv = v[0]; int bi = lane;
            #pragma unroll
            for (int j = 1; j < 4; ++j) {
                if (v[j] > bv) { bv = v[j]; bi = lane + 32 * j; }
            }
            #pragma unroll
            for (int off = 16; off > 0; off >>= 1) {
                float ov = __shfl_xor(bv, off, 32);
                int   oi = __shfl_xor(bi, off, 32);
                if (ov > bv || (ov == bv && oi < bi)) { bv = ov; bi = oi; }
            }
            if (lane == it) { myVal = bv; myIdx = bi; }           // lane L holds rank L
            if ((bi & 31) == lane) v[bi >> 5] = -3.402823466e38f; // sink below NEG
        }
        // warp softmax over the 25 ranked values
        float m = __shfl(myVal, 0, 32);                  // rank-0 value = max
        float p = (lane < TOPK) ? __expf(myVal - m) : 0.0f;
        float s = p;
        #pragma unroll
        for (int off = 16; off > 0; off >>= 1) s += __shfl_xor(s, off, 32);
        probs_s[lane] = p / s;
        besti_s[lane] = myIdx;
    }
    __syncthreads();

    // ---- phase 3: fcs[d] = sum_t probs[t] * rawTok[idx_t, d], scaled by B_diag1 ----
    for (int d = tid; d < DDIM; d += 256) {
        float acc = 0.0f;
        for (int t = 0; t < TOPK; ++t)
            acc += probs_s[t] * tokN[(size_t)besti_s[t] * DDIM + d];
        fcs_s[d] = acc * Bd1[d];
    }
    __syncthreads();

    // ---- phase 4: vals[c] = sum_d emb[c,d] * fcs[d] (8 lanes per candidate) ----
    {
        const int c = tid >> 3, sub = tid & 7;
        float acc = 0.0f;
        if (c < NC)
            for (int d = sub; d < DDIM; d += 8)
                acc += emb_s[c * LSTR + d] * fcs_s[d];
        #pragma unroll
        for (int off = 4; off > 0; off >>= 1)
            acc += __shfl_down(acc, off, 8);
        if (c < NC && sub == 0)
            out[(size_t)n * NC + c] = acc;
    }
}

extern "C" void kernel_launch(void* const* d_in, const int* in_sizes, int n_in,
                              void* d_out, int out_size, void* d_ws, size_t ws_size,
                              hipStream_t stream) {
    (void)in_sizes; (void)n_in; (void)out_size; (void)d_ws; (void)ws_size;
    const float*         emb = (const float*)d_in[0];          // (N, NC, D) f32
    const float*         tok = (const float*)d_in[1];          // (N, WIN, D) f32
    const unsigned char* msk = (const unsigned char*)d_in[2];  // (N, WIN) bool
    const float*         bd1 = (const float*)d_in[3];          // (D,)
    const float*         bd2 = (const float*)d_in[4];          // (D,)
    float*               out = (float*)d_out;                  // (N, NC)

    mention_fused_kernel<<<NMENT, 256, 0, stream>>>(emb, tok, msk, bd1, bd2, out);
}